// _VIDOBJDET_14405320311596
// MI455X (gfx1250) — compile-verified
//
#include <hip/hip_runtime.h>
#include <hip/hip_bf16.h>

// ---- problem constants (match reference) ----
#define T_WIN  16
#define RROI   512
#define DFEAT  4096
#define NCLS   31
#define NBOX   124          // 4*NCLS
#define NPAIR  15           // T-1
#define N_OUT  (T_WIN*RROI) // 8192

typedef __attribute__((ext_vector_type(2))) float v2f;
typedef __attribute__((ext_vector_type(8))) float v8f;

__device__ __forceinline__ v8f wmma_f32_k4(v2f a, v2f b, v8f c) {
  // V_WMMA_F32_16X16X4_F32 : D = A(16x4 f32) x B(4x16 f32) + C(16x16 f32)
  return __builtin_amdgcn_wmma_f32_16x16x4_f32(false, a, false, b, (short)0, c,
                                               false, false);
}

// ---------------------------------------------------------------------------
// 0) zero the packed (val|idx) atomic-max buffer
// ---------------------------------------------------------------------------
__global__ __launch_bounds__(256) void zero_u64_kernel(unsigned long long* p, int n) {
  int i = blockIdx.x * 256 + threadIdx.x;
  if (i < n) p[i] = 0ull;
}

// ---------------------------------------------------------------------------
// 1) per-row squared norms: norms[t*R+r] = sum_d feat[t,r,d]^2
// ---------------------------------------------------------------------------
__global__ __launch_bounds__(256) void norms_kernel(const float* __restrict__ feat,
                                                    float* __restrict__ norms) {
  const int wave = threadIdx.x >> 5;
  const int lane = threadIdx.x & 31;
  const int row  = blockIdx.x * 8 + wave;            // 1024 blocks * 8 waves = 8192 rows
  const float* f = feat + (size_t)row * DFEAT;
  float s = 0.f;
  #pragma unroll 4
  for (int d = lane; d < DFEAT; d += 32) { float v = f[d]; s += v * v; }
  #pragma unroll
  for (int o = 16; o; o >>= 1) s += __shfl_xor(s, o, 32);
  if (lane == 0) norms[row] = s;
}

// ---------------------------------------------------------------------------
// 2) pair matching: Gram via f32 WMMA + IoU + 1/(1+dist), column argmax via
//    packed-u64 atomicMax.  One wave computes a 64(M) x 64(N) tile:
//    16 accumulators, 8 b64 loads feed 16 WMMAs (16 flops/byte from L2).
//    grid = (8 /*tj*/, 8 /*ti*/, 15 /*pair*/), block = 32 (one wave).
// ---------------------------------------------------------------------------
__global__ __launch_bounds__(32) void pair_match_kernel(
    const float* __restrict__ feat,                  // [16,512,4096]
    const float* __restrict__ rois,                  // [16,512,4]
    const float* __restrict__ norms,                 // [16,512]
    unsigned long long* __restrict__ packed)         // [15,512]
{
  const int tj = blockIdx.x;          // column tile: j0 = tj*64
  const int ti = blockIdx.y;          // row tile:    i0 = ti*64
  const int t  = blockIdx.z;          // pair index
  const int lane = threadIdx.x;
  const int lm   = lane & 15;
  const int half = lane >> 4;

  const float* fa = feat + (size_t)t       * RROI * DFEAT;
  const float* fb = feat + (size_t)(t + 1) * RROI * DFEAT;

  // per-lane stream bases: 4 A rows (M strips) + 4 B rows (N strips)
  const float* aptr = fa + (size_t)(ti * 64 + lm) * DFEAT + half * 2;
  const float* bptr = fb + (size_t)(tj * 64 + lm) * DFEAT + half * 2;

  v8f acc[4][4];
  #pragma unroll
  for (int c = 0; c < 4; ++c)
    #pragma unroll
    for (int cj = 0; cj < 4; ++cj) acc[c][cj] = (v8f){};

  for (int kb = 0; kb < DFEAT; kb += 128) {
    // warm L2/WGP$ for the next 128-float K-block of all 8 streams
    if (kb + 128 < DFEAT) {
      #pragma unroll
      for (int c = 0; c < 4; ++c) {
        __builtin_prefetch(aptr + (size_t)(c * 16) * DFEAT + kb + 128, 0, 1);
        __builtin_prefetch(bptr + (size_t)(c * 16) * DFEAT + kb + 128, 0, 1);
      }
    }
    #pragma unroll 8
    for (int k = kb; k < kb + 128; k += 4) {
      v2f b0 = *(const v2f*)(bptr + k);
      v2f b1 = *(const v2f*)(bptr + (size_t)16 * DFEAT + k);
      v2f b2 = *(const v2f*)(bptr + (size_t)32 * DFEAT + k);
      v2f b3 = *(const v2f*)(bptr + (size_t)48 * DFEAT + k);
      v2f a0 = *(const v2f*)(aptr + k);
      v2f a1 = *(const v2f*)(aptr + (size_t)16 * DFEAT + k);
      v2f a2 = *(const v2f*)(aptr + (size_t)32 * DFEAT + k);
      v2f a3 = *(const v2f*)(aptr + (size_t)48 * DFEAT + k);
      acc[0][0] = wmma_f32_k4(a0, b0, acc[0][0]);
      acc[1][0] = wmma_f32_k4(a1, b0, acc[1][0]);
      acc[2][0] = wmma_f32_k4(a2, b0, acc[2][0]);
      acc[3][0] = wmma_f32_k4(a3, b0, acc[3][0]);
      acc[0][1] = wmma_f32_k4(a0, b1, acc[0][1]);
      acc[1][1] = wmma_f32_k4(a1, b1, acc[1][1]);
      acc[2][1] = wmma_f32_k4(a2, b1, acc[2][1]);
      acc[3][1] = wmma_f32_k4(a3, b1, acc[3][1]);
      acc[0][2] = wmma_f32_k4(a0, b2, acc[0][2]);
      acc[1][2] = wmma_f32_k4(a1, b2, acc[1][2]);
      acc[2][2] = wmma_f32_k4(a2, b2, acc[2][2]);
      acc[3][2] = wmma_f32_k4(a3, b2, acc[3][2]);
      acc[0][3] = wmma_f32_k4(a0, b3, acc[0][3]);
      acc[1][3] = wmma_f32_k4(a1, b3, acc[1][3]);
      acc[2][3] = wmma_f32_k4(a2, b3, acc[2][3]);
      acc[3][3] = wmma_f32_k4(a3, b3, acc[3][3]);
    }
  }

  // epilogue: metric + per-column argmax, one column tile (cj) at a time
  #pragma unroll
  for (int cj = 0; cj < 4; ++cj) {
    const int j = tj * 64 + cj * 16 + lm;            // this lane's column
    const float  b2v = norms[(t + 1) * RROI + j];
    const float4 rb  = *(const float4*)(rois + ((size_t)(t + 1) * RROI + j) * 4);
    const float  area_b = (rb.z - rb.x + 1.f) * (rb.w - rb.y + 1.f);

    float bestv = -1.f;
    int   besti = 0;
    #pragma unroll
    for (int c = 0; c < 4; ++c) {
      #pragma unroll
      for (int v = 0; v < 8; ++v) {
        const int i = ti * 64 + c * 16 + half * 8 + v;  // global first-frame roi
        const float dot = acc[c][cj][v];
        float d2 = norms[t * RROI + i] + b2v - 2.f * dot;
        float rev = 1.f / (1.f + __builtin_sqrtf(fmaxf(d2, 0.f)));
        const float4 ra = *(const float4*)(rois + ((size_t)t * RROI + i) * 4);
        float area_a = (ra.z - ra.x + 1.f) * (ra.w - ra.y + 1.f);
        float iw = fmaxf(fminf(ra.z, rb.z) - fmaxf(ra.x, rb.x) + 1.f, 0.f);
        float ih = fmaxf(fminf(ra.w, rb.w) - fmaxf(ra.y, rb.y) + 1.f, 0.f);
        float inter = iw * ih;
        float iou = inter / (area_a + area_b - inter);
        float m = iou + rev;
        if (m > bestv) { bestv = m; besti = i; }  // i strictly increases -> first-max kept
      }
    }

    // pack (value | 511-index): monotone for positive metrics; ties pick smaller i
    unsigned long long pk =
        ((unsigned long long)__float_as_uint(bestv) << 32) |
        (unsigned long long)(unsigned)(511 - besti);
    unsigned long long other = __shfl_xor(pk, 16, 32);  // combine both M-halves (same j)
    pk = pk > other ? pk : other;
    if (half == 0) atomicMax(&packed[t * RROI + j], pk);
  }
}

// ---------------------------------------------------------------------------
// 3) tube linking: threshold AND, prefix-sum compaction, labels.
//    single block of 512 threads (one per roi).
// ---------------------------------------------------------------------------
__global__ __launch_bounds__(512) void link_kernel(
    const unsigned long long* __restrict__ packed,   // [15,512]
    const int* __restrict__ rois_label,              // [16,512]
    int* __restrict__ idxs,                          // [15,512] out
    int* __restrict__ flags,                         // [512] out
    int* __restrict__ pos,                           // [512] out
    float* __restrict__ lbl_out)                     // d_out + 8192*155, 8192 entries
{
  __shared__ int sc[RROI];
  const int r = threadIdx.x;

  bool complete = true;
  #pragma unroll
  for (int t = 0; t < NPAIR; ++t) {
    unsigned long long p = packed[t * RROI + r];
    float v  = __uint_as_float((unsigned)(p >> 32));
    int   ix = 511 - (int)(p & 0xFFFFFFFFull);
    idxs[t * RROI + r] = ix;
    complete = complete && (v >= 1.0f);
  }
  flags[r] = complete ? 1 : 0;
  sc[r]    = complete ? 1 : 0;
  __syncthreads();

  // Hillis-Steele inclusive scan over 512
  for (int off = 1; off < RROI; off <<= 1) {
    int v   = sc[r];
    int add = (r >= off) ? sc[r - off] : 0;
    __syncthreads();
    sc[r] = v + add;
    __syncthreads();
  }
  const int p = sc[r] - 1;
  pos[r] = p;

  // tube label
  int sum = 0;
  #pragma unroll
  for (int t = 0; t < T_WIN; ++t) sum += rois_label[t * RROI + r];
  int ref = rois_label[r];
  int lbl = ((sum / T_WIN) == ref) ? ref : 0;

  // zero whole label region, then scatter complete tubes
  for (int q = r; q < N_OUT; q += RROI) lbl_out[q] = 0.f;
  __syncthreads();
  if (complete) lbl_out[p] = (float)lbl;
}

// ---------------------------------------------------------------------------
// 4a) zero compacted tube buffer [512,4096]
// ---------------------------------------------------------------------------
__global__ __launch_bounds__(256) void tube_zero_kernel(float* __restrict__ tubes) {
  size_t i = (size_t)blockIdx.x * 256 + threadIdx.x;   // grid = 8192 blocks
  tubes[i] = 0.f;
}

// ---------------------------------------------------------------------------
// 4b) gather-sum tube features into compact slots. grid = 512 blocks x 256.
// ---------------------------------------------------------------------------
__global__ __launch_bounds__(256) void tube_build_kernel(
    const float* __restrict__ feat,                  // [16,512,4096]
    const int* __restrict__ idxs,                    // [15,512]
    const int* __restrict__ flags,                   // [512]
    const int* __restrict__ pos,                     // [512]
    float* __restrict__ tubes)                       // [512,4096]
{
  const int r = blockIdx.x;
  if (!flags[r]) return;
  const int s = pos[r];
  int ji[NPAIR];
  #pragma unroll
  for (int t = 0; t < NPAIR; ++t) ji[t] = idxs[t * RROI + r];  // uniform per block

  for (int d = threadIdx.x; d < DFEAT; d += 256) {
    float a = feat[(size_t)r * DFEAT + d];
    #pragma unroll
    for (int t = 0; t < NPAIR; ++t)
      a += feat[((size_t)(t + 1) * RROI + ji[t]) * DFEAT + d];
    tubes[(size_t)s * DFEAT + d] = a;
  }
}

// ---------------------------------------------------------------------------
// 5) head GEMM: tubes[512,4096] @ W[4096,N]  via f32 WMMA.
//    grid = (8 /*nj*/, 32 /*mi*/, 2 /*cls,bbox*/), block = 32.
//    Scls is [512,32], Sbb is [512,128] (padded ld; OOB cols masked to 0).
// ---------------------------------------------------------------------------
__global__ __launch_bounds__(32) void head_gemm_kernel(
    const float* __restrict__ tubes,
    const float* __restrict__ Wcls,                  // [4096,31]
    const float* __restrict__ Wbb,                   // [4096,124]
    float* __restrict__ Scls,
    float* __restrict__ Sbb)
{
  const int tj  = blockIdx.x;
  const int tm  = blockIdx.y;
  const int mat = blockIdx.z;
  if (mat == 0 && tj >= 2) return;                    // cls only needs 32 padded cols

  const float* W = mat ? Wbb : Wcls;
  const int    N = mat ? NBOX : NCLS;
  float*       S = mat ? Sbb  : Scls;
  const int  ldS = mat ? 128  : 32;

  const int lane = threadIdx.x;
  const int lm   = lane & 15;
  const int half = lane >> 4;

  const int   col  = tj * 16 + lm;
  const int   colc = col < N ? col : (N - 1);
  const float cmsk = col < N ? 1.f : 0.f;

  const float* aptr = tubes + (size_t)(tm * 16 + lm) * DFEAT + half * 2;

  v8f acc = (v8f){};
  for (int k = 0; k < DFEAT; k += 4) {
    v2f a = *(const v2f*)(aptr + k);
    const int kk = k + half * 2;
    v2f b;
    b.x = W[(size_t)kk * N + colc] * cmsk;
    b.y = W[(size_t)(kk + 1) * N + colc] * cmsk;
    acc = wmma_f32_k4(a, b, acc);
  }

  #pragma unroll
  for (int v = 0; v < 8; ++v)
    S[(size_t)(tm * 16 + half * 8 + v) * ldS + tj * 16 + lm] = acc[v];
}

// ---------------------------------------------------------------------------
// 6) write-out: softmax over 31 cls scores + bbox, one wave per output row.
//    Rows >= 512 are zero-feature rows -> score = bias only.
// ---------------------------------------------------------------------------
__global__ __launch_bounds__(32) void writeout_kernel(
    const float* __restrict__ Scls, const float* __restrict__ Sbb,
    const float* __restrict__ bcls, const float* __restrict__ bbb,
    float* __restrict__ cls_out, float* __restrict__ bbox_out)
{
  const int q    = blockIdx.x;       // 0..8191
  const int lane = threadIdx.x;

  float sc = (lane < NCLS) ? bcls[lane] : 0.f;
  if (q < RROI) sc += Scls[(size_t)q * 32 + lane];    // padded cols are 0

  float x  = (lane < NCLS) ? sc : -3.0e38f;
  float mx = x;
  #pragma unroll
  for (int o = 16; o; o >>= 1) mx = fmaxf(mx, __shfl_xor(mx, o, 32));
  float e = (lane < NCLS) ? __expf(x - mx) : 0.f;
  float s = e;
  #pragma unroll
  for (int o = 16; o; o >>= 1) s += __shfl_xor(s, o, 32);
  if (lane < NCLS) cls_out[(size_t)q * NCLS + lane] = e / s;

  #pragma unroll
  for (int p = 0; p < 4; ++p) {
    const int col = lane + 32 * p;
    if (col < NBOX) {
      float v = bbb[col];
      if (q < RROI) v += Sbb[(size_t)q * 128 + col];
      bbox_out[(size_t)q * NBOX + col] = v;
    }
  }
}

// ---------------------------------------------------------------------------
extern "C" void kernel_launch(void* const* d_in, const int* in_sizes, int n_in,
                              void* d_out, int out_size, void* d_ws, size_t ws_size,
                              hipStream_t stream) {
  const float* feat       = (const float*)d_in[0];   // [16,512,4096]
  const float* rois       = (const float*)d_in[1];   // [16,512,4]
  const int*   rois_label = (const int*)  d_in[2];   // [16,512]
  const float* Wcls       = (const float*)d_in[3];   // [4096,31]
  const float* bcls       = (const float*)d_in[4];   // [31]
  const float* Wbb        = (const float*)d_in[5];   // [4096,124]
  const float* bbb        = (const float*)d_in[6];   // [124]

  float* out      = (float*)d_out;
  float* cls_out  = out;                              // 8192*31
  float* bbox_out = out + (size_t)N_OUT * NCLS;       // 8192*124
  float* lbl_out  = out + (size_t)N_OUT * (NCLS + NBOX); // 8192

  // workspace carve-up
  char* ws = (char*)d_ws;
  size_t off = 0;
  float* norms = (float*)(ws + off);               off += (size_t)T_WIN * RROI * 4;     // 32 KB
  unsigned long long* packed =
      (unsigned long long*)(ws + off);             off += (size_t)NPAIR * RROI * 8;     // 60 KB
  int* idxs  = (int*)(ws + off);                   off += (size_t)NPAIR * RROI * 4;     // 30 KB
  int* flags = (int*)(ws + off);                   off += (size_t)RROI * 4;
  int* pos   = (int*)(ws + off);                   off += (size_t)RROI * 4;
  off = (off + 255) & ~(size_t)255;
  float* tubes = (float*)(ws + off);               off += (size_t)RROI * DFEAT * 4;     // 8 MB
  float* Scls  = (float*)(ws + off);               off += (size_t)RROI * 32 * 4;
  float* Sbb   = (float*)(ws + off);               off += (size_t)RROI * 128 * 4;

  // 0) init atomic-max buffer
  zero_u64_kernel<<<(NPAIR * RROI + 255) / 256, 256, 0, stream>>>(packed, NPAIR * RROI);
  // 1) squared norms
  norms_kernel<<<(T_WIN * RROI) / 8, 256, 0, stream>>>(feat, norms);
  // 2) pair matching (WMMA Gram + IoU + argmax), 64x64 tile per wave
  pair_match_kernel<<<dim3(8, 8, NPAIR), 32, 0, stream>>>(feat, rois, norms, packed);
  // 3) linking / compaction / labels
  link_kernel<<<1, RROI, 0, stream>>>(packed, rois_label, idxs, flags, pos, lbl_out);
  // 4) tube feature build
  tube_zero_kernel<<<(RROI * DFEAT) / 256, 256, 0, stream>>>(tubes);
  tube_build_kernel<<<RROI, 256, 0, stream>>>(feat, idxs, flags, pos, tubes);
  // 5) head GEMM (WMMA)
  head_gemm_kernel<<<dim3(8, 32, 2), 32, 0, stream>>>(tubes, Wcls, Wbb, Scls, Sbb);
  // 6) softmax + bbox write-out
  writeout_kernel<<<N_OUT, 32, 0, stream>>>(Scls, Sbb, bcls, bbb, cls_out, bbox_out);
}